// SlotAttentionVariant_84353157693442
// MI455X (gfx1250) — compile-verified
//
#include <hip/hip_runtime.h>
#include <math.h>

typedef float v2f __attribute__((ext_vector_type(2)));
typedef float v8f __attribute__((ext_vector_type(8)));
typedef unsigned int v4u __attribute__((ext_vector_type(4)));
typedef int v8i __attribute__((ext_vector_type(8)));
typedef int v4i __attribute__((ext_vector_type(4)));

#define B_   64
#define N_   4096
#define DIN_ 256
#define S_   8
#define D_   64
#define DM_  128
#define LN_EPS 1e-5f
#define SCALE_ 0.125f   // D^-0.5

// ------------------------- Phase 1: LN + K/V projection -------------------------
#define P1_ROWS 64      // rows of x per block  (LDS-capacity probe: ~84KB static)
#define P1_THREADS 128  // 4 waves, 16 rows each
#define XSTR 260        // padded LDS row stride for x_ln (bank-conflict-free A frags)
#define WCHK 32         // K-chunk of Wk/Wv staged in LDS
#define WSTR 68         // padded row stride for weight chunks (16B-aligned rows)

__global__ __launch_bounds__(P1_THREADS)
void ln_kv_proj(const float* __restrict__ inputs,
                const float* __restrict__ ln_g, const float* __restrict__ ln_b,
                const float* __restrict__ Wk,  const float* __restrict__ Wv,
                float* __restrict__ kout, float* __restrict__ vout)
{
  __shared__ float xln[P1_ROWS][XSTR];   // 66,560 B
  __shared__ float wkc[WCHK][WSTR];      //  8,704 B
  __shared__ float wvc[WCHK][WSTR];      //  8,704 B

  const int tid  = threadIdx.x;
  const int wave = tid >> 5;
  const int lane = tid & 31;
  const size_t rowbase = (size_t)blockIdx.x * P1_ROWS;

  // ---- LayerNorm: each wave normalizes 16 rows of DIN=256 ----
  float gv[8], bv[8];
  #pragma unroll
  for (int j = 0; j < 8; ++j) { gv[j] = ln_g[lane + j*32]; bv[j] = ln_b[lane + j*32]; }

  for (int r = wave*16; r < wave*16 + 16; ++r) {
    const float* xr = inputs + (rowbase + r) * DIN_;
    float vals[8];
    float s = 0.f;
    #pragma unroll
    for (int j = 0; j < 8; ++j) { vals[j] = xr[lane + j*32]; s += vals[j]; }
    #pragma unroll
    for (int m = 16; m >= 1; m >>= 1) s += __shfl_xor(s, m, 32);
    const float mean = s * (1.0f / DIN_);
    float vs = 0.f;
    #pragma unroll
    for (int j = 0; j < 8; ++j) { float d0 = vals[j] - mean; vs += d0 * d0; }
    #pragma unroll
    for (int m = 16; m >= 1; m >>= 1) vs += __shfl_xor(vs, m, 32);
    const float rstd = rsqrtf(vs * (1.0f / DIN_) + LN_EPS);
    #pragma unroll
    for (int j = 0; j < 8; ++j)
      xln[r][lane + j*32] = (vals[j] - mean) * rstd * gv[j] + bv[j];
  }
  __syncthreads();

  // ---- WMMA GEMM: k = xln @ Wk, v = xln @ Wv  (M=16/wave, N=64, K=256) ----
  v8f acck[4] = {};
  v8f accv[4] = {};

  const int rbl    = wave * 16;
  const int mlo    = lane & 15;
  const int mrow   = rbl + mlo;
  const int khalf  = (lane < 16) ? 0 : 2;

  for (int kc = 0; kc < DIN_; kc += WCHK) {
    // stage Wk/Wv chunk (coalesced float4)
    for (int i4 = tid; i4 < (WCHK * D_) / 4; i4 += P1_THREADS) {
      int fl = i4 * 4;
      int kr = fl >> 6;
      int cc = fl & 63;
      *(float4*)&wkc[kr][cc] = *(const float4*)&Wk[(kc + kr) * D_ + cc];
      *(float4*)&wvc[kr][cc] = *(const float4*)&Wv[(kc + kr) * D_ + cc];
    }
    __syncthreads();

    #pragma unroll
    for (int ks = 0; ks < WCHK; ks += 4) {
      v2f a;
      a.x = xln[mrow][kc + ks + khalf];
      a.y = xln[mrow][kc + ks + khalf + 1];
      #pragma unroll
      for (int nt = 0; nt < 4; ++nt) {
        const int nn = nt * 16 + mlo;
        v2f bk, bv2;
        bk.x  = wkc[ks + khalf][nn];
        bk.y  = wkc[ks + khalf + 1][nn];
        bv2.x = wvc[ks + khalf][nn];
        bv2.y = wvc[ks + khalf + 1][nn];
        acck[nt] = __builtin_amdgcn_wmma_f32_16x16x4_f32(
            false, a, false, bk,  (short)0, acck[nt], false, false);
        accv[nt] = __builtin_amdgcn_wmma_f32_16x16x4_f32(
            false, a, false, bv2, (short)0, accv[nt], false, false);
      }
    }
    __syncthreads();
  }

  // ---- Store k/v tiles (C layout: VGPR j -> rows j / j+8; N = lane&15) ----
  const int mAdd = (lane < 16) ? 0 : 8;
  #pragma unroll
  for (int nt = 0; nt < 4; ++nt) {
    #pragma unroll
    for (int j = 0; j < 8; ++j) {
      size_t row = rowbase + rbl + j + mAdd;
      int col = nt * 16 + mlo;
      kout[row * D_ + col] = acck[nt][j];
      vout[row * D_ + col] = accv[nt][j];
    }
  }
}

// ------------------------- Phase 2: slot-attention iterations -------------------------
#define CH 32          // k/v rows per LDS chunk (double-buffered via TDM)
#define NCHUNK (N_ / CH)
#define KSTR 65        // padded stride: (n*65+d)%64 conflict-free; matches TDM pad

// Issue one TDM 2-D tile load: CH x 64 f32 tile, row stride 64, with 1-DWORD pad
// inserted after every 64 DWORDs (=> LDS stride 65). CDNA5 ISA.
__device__ __forceinline__ void tdm_load_tile(unsigned lds_addr, const float* gptr) {
  unsigned long long ga = (unsigned long long)(size_t)gptr;
  v4u g0;
  g0[0] = 1u;                                               // count=1, user mode
  g0[1] = lds_addr;                                         // LDS dest byte address
  g0[2] = (unsigned)(ga & 0xFFFFFFFFu);                     // global_addr[31:0]
  g0[3] = (unsigned)((ga >> 32) & 0x1FFFFFFu) | (2u << 30); // global_addr[56:32] | type=2
  v8i g1;
  g1[0] = (int)((2u << 16)      // data_size = 4B
              | (1u << 20)      // pad_enable
              | (5u << 22)      // pad_interval: 64 DWORDs
              | (0u << 25));    // pad_amount: 1 DWORD
  g1[1] = (int)(64u << 16);     // tensor_dim0 = 64 (atomic_barrier_addr = 0)
  g1[2] = (int)(4096u << 16);   // tensor_dim1 = 4096
  g1[3] = (int)(64u << 16);     // tile_dim0 = 64
  g1[4] = (int)CH;              // tile_dim1 = CH, tile_dim2 = 0
  g1[5] = (int)64u;             // tensor_dim0_stride = 64
  g1[6] = 0;
  g1[7] = 0;
  v4i z4 = {0, 0, 0, 0};
  v8i z8 = {0, 0, 0, 0, 0, 0, 0, 0};
  // amdgpu-toolchain (clang-23) 6-arg form: (g0, g1, g2, g3, g4, cpol)
  __builtin_amdgcn_tensor_load_to_lds(g0, g1, z4, z4, z8, 0);
}

__device__ __forceinline__ float dot64(const float* __restrict__ a,
                                       const float* __restrict__ w) {
  float acc = 0.f;
  #pragma unroll 8
  for (int d = 0; d < D_; ++d) acc += a[d] * w[d];
  return acc;
}

__global__ __launch_bounds__(256)
void slot_attn_iter(const float* __restrict__ kin, const float* __restrict__ vin,
                    const float* __restrict__ slots_init,
                    const float* __restrict__ ln_s_g, const float* __restrict__ ln_s_b,
                    const float* __restrict__ ln_m_g, const float* __restrict__ ln_m_b,
                    const float* __restrict__ Wq,
                    const float* __restrict__ w_ih, const float* __restrict__ w_hh,
                    const float* __restrict__ b_ih, const float* __restrict__ b_hh,
                    const float* __restrict__ mlp_w1, const float* __restrict__ mlp_b1,
                    const float* __restrict__ mlp_w2, const float* __restrict__ mlp_b2,
                    const float* __restrict__ mu, const float* __restrict__ lsig,
                    float* __restrict__ out)
{
  __shared__ float kcb[2][CH][KSTR];   // 16,640 B  (ping-pong)
  __shared__ float vcb[2][CH][KSTR];   // 16,640 B
  __shared__ float attn[CH][S_];       //  1,024 B
  __shared__ float slots[S_][D_];
  __shared__ float prev[S_][D_];
  __shared__ float q[S_][D_];
  __shared__ float lnb[S_][D_];
  __shared__ float updb[S_][D_];
  __shared__ float hbuf[S_][DM_];      //  4,096 B
  __shared__ float red[CH][S_];        //  1,024 B
  __shared__ float colsum[S_];

  const int tid = threadIdx.x;
  const int b   = blockIdx.x;
  const float* kb = kin + (size_t)b * N_ * D_;
  const float* vb = vin + (size_t)b * N_ * D_;

  const unsigned klds[2] = { (unsigned)(size_t)&kcb[0][0][0],
                             (unsigned)(size_t)&kcb[1][0][0] };
  const unsigned vlds[2] = { (unsigned)(size_t)&vcb[0][0][0],
                             (unsigned)(size_t)&vcb[1][0][0] };

  // element mapping for (s,d) grids of 512 elems: e0 = tid, e1 = tid + 256
  const int s0 = tid >> 6;         // 0..3
  const int d0 = tid & 63;
  const int s1 = s0 + 4;           // 4..7

  // ---- init slots = mu + exp(log_sigma) * slots_init ----
  {
    const float* si = slots_init + (size_t)b * S_ * D_;
    float e = expf(lsig[d0]);
    slots[s0][d0] = mu[d0] + e * si[s0 * D_ + d0];
    slots[s1][d0] = mu[d0] + e * si[s1 * D_ + d0];
  }
  __syncthreads();

  for (int it = 0; it < 3; ++it) {
    // slots_prev = slots
    prev[s0][d0] = slots[s0][d0];
    prev[s1][d0] = slots[s1][d0];
    __syncthreads();

    // ---- LN_s(slots) -> lnb ----
    if (tid < S_) {
      float m = 0.f;
      for (int d = 0; d < D_; ++d) m += slots[tid][d];
      m *= (1.0f / D_);
      float var = 0.f;
      for (int d = 0; d < D_; ++d) { float df = slots[tid][d] - m; var += df * df; }
      float rs = rsqrtf(var * (1.0f / D_) + LN_EPS);
      for (int d = 0; d < D_; ++d)
        lnb[tid][d] = (slots[tid][d] - m) * rs * ln_s_g[d] + ln_s_b[d];
    }
    __syncthreads();

    // ---- q = lnb @ Wq ----
    {
      float a0 = 0.f, a1 = 0.f;
      for (int d = 0; d < D_; ++d) {
        float w = Wq[d * D_ + d0];
        a0 += lnb[s0][d] * w;
        a1 += lnb[s1][d] * w;
      }
      q[s0][d0] = a0;
      q[s1][d0] = a1;
    }

    // ---- prologue: TDM-load chunk 0 into buffer 0 (wave 0 issues) ----
    if (tid < 32) {
      tdm_load_tile(klds[0], kb);
      tdm_load_tile(vlds[0], vb);
    }
    __syncthreads();

    float accu0 = 0.f, accu1 = 0.f;   // updates accumulators (s0,d0),(s1,d0)
    float csum[S_];
    #pragma unroll
    for (int s = 0; s < S_; ++s) csum[s] = 0.f;

    const int nrow  = tid & 31;      // cost-pass row
    const int srow  = tid >> 5;      // cost-pass slot (0..7)

    for (int ch = 0; ch < NCHUNK; ++ch) {
      const int cur = ch & 1;
      // issue next chunk into the other buffer, then wait for current chunk
      if (tid < 32) {
        if (ch + 1 < NCHUNK) {
          const size_t nb = (size_t)(ch + 1) * CH * D_;
          tdm_load_tile(klds[1 - cur], kb + nb);
          tdm_load_tile(vlds[1 - cur], vb + nb);
          __builtin_amdgcn_s_wait_tensorcnt(2);   // current pair complete
        } else {
          __builtin_amdgcn_s_wait_tensorcnt(0);
        }
      }
      __syncthreads();

      // cost: thread (nrow, srow) computes one slot dot
      {
        float c = 0.f;
        #pragma unroll 8
        for (int d = 0; d < D_; ++d) c += kcb[cur][nrow][d] * q[srow][d];
        attn[nrow][srow] = SCALE_ * c;
      }
      __syncthreads();

      // softmax over S per row + 1e-8 (row owners: tid < CH)
      if (tid < CH) {
        float cvals[S_];
        float mx = -3.4e38f;
        #pragma unroll
        for (int s = 0; s < S_; ++s) { cvals[s] = attn[tid][s]; mx = fmaxf(mx, cvals[s]); }
        float sum = 0.f;
        #pragma unroll
        for (int s = 0; s < S_; ++s) { cvals[s] = expf(cvals[s] - mx); sum += cvals[s]; }
        float inv = 1.0f / sum;
        #pragma unroll
        for (int s = 0; s < S_; ++s) {
          float a = cvals[s] * inv + 1e-8f;
          attn[tid][s] = a;
          csum[s] += a;
        }
      }
      __syncthreads();

      // updates partial: upd[s][d] += sum_n attn[n][s] * v[n][d]
      #pragma unroll 8
      for (int n = 0; n < CH; ++n) {
        accu0 += attn[n][s0] * vcb[cur][n][d0];
        accu1 += attn[n][s1] * vcb[cur][n][d0];
      }
      __syncthreads();
    }

    // ---- reduce attention column sums over the CH row-owner threads ----
    if (tid < CH) {
      #pragma unroll
      for (int s = 0; s < S_; ++s) red[tid][s] = csum[s];
    }
    __syncthreads();
    if (tid < S_) {
      float t = 0.f;
      for (int i = 0; i < CH; ++i) t += red[i][tid];
      colsum[tid] = t;
    }
    __syncthreads();

    // ---- normalized updates into LDS ----
    updb[s0][d0] = accu0 / colsum[s0];
    updb[s1][d0] = accu1 / colsum[s1];
    __syncthreads();

    // ---- GRUCell(updates, slots_prev) -> slots ----
    #pragma unroll
    for (int e = 0; e < 2; ++e) {
      const int s = (e == 0) ? s0 : s1;
      const float* u = &updb[s][0];
      const float* p = &prev[s][0];
      float xr = dot64(u, w_ih + (size_t)d0 * D_)          + b_ih[d0];
      float hr = dot64(p, w_hh + (size_t)d0 * D_)          + b_hh[d0];
      float xz = dot64(u, w_ih + (size_t)(D_ + d0) * D_)   + b_ih[D_ + d0];
      float hz = dot64(p, w_hh + (size_t)(D_ + d0) * D_)   + b_hh[D_ + d0];
      float xn = dot64(u, w_ih + (size_t)(2*D_ + d0) * D_) + b_ih[2*D_ + d0];
      float hn = dot64(p, w_hh + (size_t)(2*D_ + d0) * D_) + b_hh[2*D_ + d0];
      float r  = 1.0f / (1.0f + expf(-(xr + hr)));
      float z  = 1.0f / (1.0f + expf(-(xz + hz)));
      float nn = tanhf(xn + r * hn);
      slots[s][d0] = (1.0f - z) * nn + z * prev[s][d0];
    }
    __syncthreads();

    // ---- LN_m(slots) -> lnb ----
    if (tid < S_) {
      float m = 0.f;
      for (int d = 0; d < D_; ++d) m += slots[tid][d];
      m *= (1.0f / D_);
      float var = 0.f;
      for (int d = 0; d < D_; ++d) { float df = slots[tid][d] - m; var += df * df; }
      float rs = rsqrtf(var * (1.0f / D_) + LN_EPS);
      for (int d = 0; d < D_; ++d)
        lnb[tid][d] = (slots[tid][d] - m) * rs * ln_m_g[d] + ln_m_b[d];
    }
    __syncthreads();

    // ---- h = relu(lnb @ mlp_w1 + b1) ----
    #pragma unroll
    for (int e = 0; e < 4; ++e) {
      int idx = tid + e * 256;
      int hs = idx >> 7, hm = idx & 127;
      float a = mlp_b1[hm];
      for (int d = 0; d < D_; ++d) a += lnb[hs][d] * mlp_w1[d * DM_ + hm];
      hbuf[hs][hm] = fmaxf(a, 0.f);
    }
    __syncthreads();

    // ---- slots += h @ mlp_w2 + b2 ----
    #pragma unroll
    for (int e = 0; e < 2; ++e) {
      const int s = (e == 0) ? s0 : s1;
      float a = mlp_b2[d0];
      for (int m = 0; m < DM_; ++m) a += hbuf[s][m] * mlp_w2[m * D_ + d0];
      slots[s][d0] += a;
    }
    __syncthreads();
  }

  // ---- write output ----
  out[((size_t)b * S_ + s0) * D_ + d0] = slots[s0][d0];
  out[((size_t)b * S_ + s1) * D_ + d0] = slots[s1][d0];
}

// ------------------------- launch -------------------------
extern "C" void kernel_launch(void* const* d_in, const int* in_sizes, int n_in,
                              void* d_out, int out_size, void* d_ws, size_t ws_size,
                              hipStream_t stream) {
  const float* inputs     = (const float*)d_in[0];
  const float* slots_init = (const float*)d_in[1];
  const float* ln_in_g    = (const float*)d_in[2];
  const float* ln_in_b    = (const float*)d_in[3];
  const float* ln_s_g     = (const float*)d_in[4];
  const float* ln_s_b     = (const float*)d_in[5];
  const float* ln_m_g     = (const float*)d_in[6];
  const float* ln_m_b     = (const float*)d_in[7];
  const float* Wq         = (const float*)d_in[8];
  const float* Wk         = (const float*)d_in[9];
  const float* Wv         = (const float*)d_in[10];
  const float* w_ih       = (const float*)d_in[11];
  const float* w_hh       = (const float*)d_in[12];
  const float* b_ih       = (const float*)d_in[13];
  const float* b_hh       = (const float*)d_in[14];
  const float* mlp_w1     = (const float*)d_in[15];
  const float* mlp_b1     = (const float*)d_in[16];
  const float* mlp_w2     = (const float*)d_in[17];
  const float* mlp_b2     = (const float*)d_in[18];
  const float* mu         = (const float*)d_in[19];
  const float* lsig       = (const float*)d_in[20];

  float* kbuf = (float*)d_ws;                       // 64 MB
  float* vbuf = kbuf + (size_t)B_ * N_ * D_;        // 64 MB
  float* out  = (float*)d_out;

  ln_kv_proj<<<dim3((B_ * N_) / P1_ROWS), dim3(P1_THREADS), 0, stream>>>(
      inputs, ln_in_g, ln_in_b, Wk, Wv, kbuf, vbuf);

  slot_attn_iter<<<dim3(B_), dim3(256), 0, stream>>>(
      kbuf, vbuf, slots_init, ln_s_g, ln_s_b, ln_m_g, ln_m_b,
      Wq, w_ih, w_hh, b_ih, b_hh, mlp_w1, mlp_b1, mlp_w2, mlp_b2,
      mu, lsig, out);
}